// DetectionOutput_72627896975855
// MI455X (gfx1250) — compile-verified
//
#include <hip/hip_runtime.h>
#include <hip/hip_bf16.h>
#include <stdint.h>

// Problem constants (from reference)
#define B_    64
#define P_    27559
#define K_    400
#define P_PAD 27648          // 1024 * 27, padded key count
#define SEG   27             // keys per thread segment
#define CONF_TH 0.01f
#define NMS_TH  0.35f

typedef __attribute__((ext_vector_type(4))) unsigned int u32x4_t;
typedef __attribute__((ext_vector_type(8))) int          i32x8_t;
typedef __attribute__((ext_vector_type(4))) int          i32x4_t;

// ---------------------------------------------------------------------------
// Kernel 1: box decode (memory-bound elementwise)
// ---------------------------------------------------------------------------
__global__ __launch_bounds__(256) void decode_kernel(
    const float4* __restrict__ locs, const float4* __restrict__ priors,
    const float4* __restrict__ vars, float4* __restrict__ out) {
  int idx = blockIdx.x * blockDim.x + threadIdx.x;
  if (idx >= B_ * P_) return;
  int p = idx % P_;
  float4 l = locs[idx], pr = priors[p], v = vars[p];
  float pcx = 0.5f * (pr.x + pr.z), pcy = 0.5f * (pr.y + pr.w);
  float pw = pr.z - pr.x, ph = pr.w - pr.y;
  float cx = v.x * l.x * pw + pcx;
  float cy = v.y * l.y * ph + pcy;
  float w  = expf(v.z * l.z) * pw;
  float h  = expf(v.w * l.w) * ph;
  float4 o;
  o.x = cx - 0.5f * w; o.y = cy - 0.5f * h;
  o.z = cx + 0.5f * w; o.w = cy + 0.5f * h;
  out[idx] = o;
}

// ---------------------------------------------------------------------------
// Kernel 2: exact top-400 per batch.
//  - TDM DMA of the full 220KB conf row into LDS (320KB/WGP on CDNA5)
//  - in-place conversion to monotonic (score_bits<<32 | ~idx) keys
//  - segmented tournament: per round, 32 wave-lane0 scans + thread0 scan +
//    27-element rescan by the winning segment's owner => 3 barriers/round
// ---------------------------------------------------------------------------
__global__ __launch_bounds__(1024) void topk_kernel(
    const float* __restrict__ conf, int* __restrict__ gidx,
    float* __restrict__ gval, float* __restrict__ out_idx) {
  extern __shared__ unsigned long long smem[];
  unsigned long long* keys   = smem;             // P_PAD keys (raw conf pairs first)
  unsigned long long* segmax = keys + P_PAD;     // 1024 per-segment maxes
  __shared__ unsigned long long wavemax[32];
  __shared__ int wavei[32];
  __shared__ int s_winseg;
  __shared__ unsigned s_winp;

  const int b = blockIdx.x;
  const int t = threadIdx.x;

  // ---- TDM: DMA conf_scores[b] (2*P_ contiguous floats) into LDS ----------
  if (t == 0) {
    uint64_t gaddr = (uint64_t)(uintptr_t)(conf + (size_t)b * (2 * P_));
    uint32_t laddr = (uint32_t)(uintptr_t)keys;   // low 32 bits = LDS byte addr
    const uint32_t n = 2u * P_;                   // 55118 4-byte elements
    u32x4_t g0;
    g0[0] = 1u;                                              // count=1, valid
    g0[1] = laddr;                                           // lds_addr
    g0[2] = (uint32_t)gaddr;                                 // global_addr lo
    g0[3] = (uint32_t)((gaddr >> 32) & 0x01FFFFFFu) | (2u << 30); // hi | type=2
    i32x8_t g1;
    g1[0] = (int)(2u << 16);                          // data_size = 4 bytes
    g1[1] = (int)((n & 0xFFFFu) << 16);               // tensor_dim0[15:0]
    g1[2] = (int)(((n >> 16) & 0xFFFFu) | (1u << 16));// dim0[31:16] | tensor_dim1=1
    g1[3] = (int)((n & 0xFFFFu) << 16);               // dim1 hi=0 | tile_dim0=n
    g1[4] = 1;                                        // tile_dim1=1, tile_dim2=0
    g1[5] = (int)n;                                   // tensor_dim0_stride lo
    g1[6] = 0;                                        // stride0 hi | stride1 lo
    g1[7] = 0;                                        // stride1 hi
    i32x4_t gz = {0, 0, 0, 0};
#if defined(__clang_major__) && __clang_major__ >= 23
    i32x8_t gz8 = {0, 0, 0, 0, 0, 0, 0, 0};
    __builtin_amdgcn_tensor_load_to_lds(g0, g1, gz, gz, gz8, 0);
#else
    __builtin_amdgcn_tensor_load_to_lds(g0, g1, gz, gz, 0);
#endif
    __builtin_amdgcn_s_wait_tensorcnt(0);
  }
  __syncthreads();

  // ---- convert (conf0,conf1) pairs in place to sort keys -------------------
  // pair occupies the same 8 bytes: low dword = conf0, high dword = conf1.
  // key = score_bits(conf1)<<32 | ~p  => max key == top_k ordering w/ ties.
  for (int p = t; p < P_PAD; p += 1024) {
    unsigned long long key = 0ull;
    if (p < P_) {
      unsigned long long v = keys[p];
      key = (v & 0xFFFFFFFF00000000ull) | (unsigned)(~(unsigned)p);
    }
    keys[p] = key;
  }
  __syncthreads();

  // ---- per-thread segment max ---------------------------------------------
  {
    unsigned long long m = 0ull;
    int base = t * SEG;
    for (int j = 0; j < SEG; ++j) {
      unsigned long long kk = keys[base + j];
      if (kk > m) m = kk;
    }
    segmax[t] = m;
  }
  __syncthreads();

  // ---- 400 tournament rounds (3 barriers per round) ------------------------
  for (int k = 0; k < K_; ++k) {
    // level 1: one lane per wave scans that wave's 32 segment-maxes
    if ((t & 31) == 0) {
      int w = t >> 5;
      unsigned long long m = 0ull;
      int mi = t;
      for (int j = 0; j < 32; ++j) {
        unsigned long long kk = segmax[t + j];
        if (kk > m) { m = kk; mi = t + j; }
      }
      wavemax[w] = m;
      wavei[w] = mi;
    }
    __syncthreads();
    // level 2: thread 0 scans the 32 wave maxes and emits the winner
    if (t == 0) {
      unsigned long long m = 0ull;
      int mi = 0;
      for (int w = 0; w < 32; ++w) {
        unsigned long long kk = wavemax[w];
        if (kk > m) { m = kk; mi = wavei[w]; }
      }
      unsigned p = ~(unsigned)m;
      gidx[b * K_ + k]    = (int)p;
      gval[b * K_ + k]    = __uint_as_float((unsigned)(m >> 32));
      out_idx[b * K_ + k] = (float)p;
      s_winseg = mi;
      s_winp   = p;
    }
    __syncthreads();
    // remove winner, owning thread rescans its 27-key segment
    if (t == s_winseg) {
      keys[s_winp] = 0ull;
      unsigned long long m = 0ull;
      int base = t * SEG;
      for (int j = 0; j < SEG; ++j) {
        unsigned long long kk = keys[base + j];
        if (kk > m) m = kk;
      }
      segmax[t] = m;
    }
    __syncthreads();
  }
}

// ---------------------------------------------------------------------------
// Kernel 3: greedy NMS, one workgroup per batch, exact reference semantics
// ---------------------------------------------------------------------------
__global__ __launch_bounds__(512) void nms_kernel(
    const float4* __restrict__ decoded, const int* __restrict__ gidx,
    const float* __restrict__ gval, int* __restrict__ gkeep,
    float* __restrict__ out_keep) {
  __shared__ float sx1[K_], sy1[K_], sx2[K_], sy2[K_], sar[K_];
  __shared__ int skeep[K_];
  const int b = blockIdx.x;
  const int j = threadIdx.x;
  float x1 = 0.f, y1 = 0.f, x2 = 0.f, y2 = 0.f, ar = 0.f;
  if (j < K_) {
    int i = gidx[b * K_ + j];
    float4 bx = decoded[(size_t)b * P_ + i];
    x1 = bx.x; y1 = bx.y; x2 = bx.z; y2 = bx.w;
    ar = (x2 - x1 + 1.f) * (y2 - y1 + 1.f);
    sx1[j] = x1; sy1[j] = y1; sx2[j] = x2; sy2[j] = y2; sar[j] = ar;
    skeep[j] = (gval[b * K_ + j] > CONF_TH) ? 1 : 0;
  }
  __syncthreads();
  for (int i = 0; i < K_; ++i) {
    if (skeep[i] && j < K_ && j > i) {
      float xx1 = fmaxf(sx1[i], x1);
      float yy1 = fmaxf(sy1[i], y1);
      float xx2 = fminf(sx2[i], x2);
      float yy2 = fminf(sy2[i], y2);
      float w = fmaxf(xx2 - xx1 + 1.f, 0.f);
      float h = fmaxf(yy2 - yy1 + 1.f, 0.f);
      float inter = w * h;
      float iou = inter / (sar[i] + ar - inter);
      if (iou > NMS_TH) skeep[j] = 0;
    }
    __syncthreads();
  }
  if (j < K_) {
    out_keep[b * K_ + j] = skeep[j] ? 1.f : 0.f;
    gkeep[b * K_ + j] = skeep[j];
  }
}

// ---------------------------------------------------------------------------
// Kernel 4: cross-batch masked gather (L2-resident; writes fully coalesced)
// out[b2][b1][k] = src[b1][ idx[b2][k] ] * keep[b2][k]
// ---------------------------------------------------------------------------
__global__ __launch_bounds__(256) void gather_kernel(
    const float4* __restrict__ decoded, const float2* __restrict__ conf,
    const float* __restrict__ state, const int* __restrict__ gidx,
    const int* __restrict__ gkeep, float4* __restrict__ sel_boxes,
    float2* __restrict__ sel_conf, float* __restrict__ sel_state) {
  int tid = blockIdx.x * blockDim.x + threadIdx.x;
  if (tid >= B_ * B_ * K_) return;
  int k  = tid % K_;
  int b1 = (tid / K_) % B_;
  int b2 = tid / (K_ * B_);
  int i  = gidx[b2 * K_ + k];
  float m = (float)gkeep[b2 * K_ + k];
  size_t src = (size_t)b1 * P_ + (size_t)i;
  size_t dst = (size_t)tid;                 // == ((b2*B_+b1)*K_ + k)
  const float* sp = state + src * 3;
  __builtin_prefetch(sp, 0, 0);             // global_prefetch_b8: start 3rd line
  float4 bx = decoded[src];
  bx.x *= m; bx.y *= m; bx.z *= m; bx.w *= m;
  sel_boxes[dst] = bx;
  float2 cf = conf[src];
  cf.x *= m; cf.y *= m;
  sel_conf[dst] = cf;
  float* dp = sel_state + dst * 3;
  dp[0] = sp[0] * m; dp[1] = sp[1] * m; dp[2] = sp[2] * m;
}

// ---------------------------------------------------------------------------
extern "C" void kernel_launch(void* const* d_in, const int* in_sizes, int n_in,
                              void* d_out, int out_size, void* d_ws, size_t ws_size,
                              hipStream_t stream) {
  const float* locs   = (const float*)d_in[0];
  const float* conf   = (const float*)d_in[1];
  const float* state  = (const float*)d_in[2];
  const float* priors = (const float*)d_in[3];
  const float* vars   = (const float*)d_in[4];

  float* out = (float*)d_out;
  size_t off = 0;
  float* o_decoded = out + off; off += (size_t)B_ * P_ * 4;       // decoded
  float* o_boxes   = out + off; off += (size_t)B_ * B_ * K_ * 4;  // sel_boxes
  float* o_conf    = out + off; off += (size_t)B_ * B_ * K_ * 2;  // sel_conf
  float* o_state   = out + off; off += (size_t)B_ * B_ * K_ * 3;  // sel_state
  float* o_keep    = out + off; off += (size_t)B_ * K_;           // keeps
  float* o_idx     = out + off;                                    // idxs

  int*   gidx  = (int*)d_ws;                 // B_*K_ selected indices
  float* gval  = (float*)(gidx + B_ * K_);   // B_*K_ selected scores
  int*   gkeep = (int*)(gval + B_ * K_);     // B_*K_ keep flags

  int bp = B_ * P_;
  decode_kernel<<<(bp + 255) / 256, 256, 0, stream>>>(
      (const float4*)locs, (const float4*)priors, (const float4*)vars,
      (float4*)o_decoded);

  size_t lds_bytes = (size_t)P_PAD * 8 + 1024 * 8;   // keys + segmax
  topk_kernel<<<B_, 1024, lds_bytes, stream>>>(conf, gidx, gval, o_idx);

  nms_kernel<<<B_, 512, 0, stream>>>((const float4*)o_decoded, gidx, gval,
                                     gkeep, o_keep);

  int tot = B_ * B_ * K_;
  gather_kernel<<<(tot + 255) / 256, 256, 0, stream>>>(
      (const float4*)o_decoded, (const float2*)conf, state, gidx, gkeep,
      (float4*)o_boxes, (float2*)o_conf, o_state);
}